// GraphSAGEModel_7413113552974
// MI455X (gfx1250) — compile-verified
//
#include <hip/hip_runtime.h>
#include <hip/hip_bf16.h>
#include <math.h>

typedef __attribute__((ext_vector_type(2))) float v2f;
typedef __attribute__((ext_vector_type(8))) float v8f;

#define N_NODES 50000
#define N_EDGES 600000
#define D_FEAT 128
#define N_CLASSES 47

// ---------------------------------------------------------------- zero fill
__global__ void zero_f32(float* __restrict__ p, size_t n) {
    size_t i = (size_t)blockIdx.x * blockDim.x + threadIdx.x;
    size_t stride = (size_t)gridDim.x * blockDim.x;
    for (; i < n; i += stride) p[i] = 0.0f;
}

// ---------------------------------------------------------------- degree
__global__ void deg_kernel(const long long* __restrict__ dst, float* __restrict__ deg, int E) {
    int e = blockIdx.x * blockDim.x + threadIdx.x;
    if (e < E) atomicAdd(&deg[(int)dst[e]], 1.0f);
}

// ---------------------------------------------------------------- scatter-add of 128-wide rows
// one thread per (edge, 4-float chunk): 32 threads per edge
__global__ void scatter_add(const float* __restrict__ x, const long long* __restrict__ src,
                            const long long* __restrict__ dst, float* __restrict__ agg, int E) {
    long long t = (long long)blockIdx.x * blockDim.x + threadIdx.x;
    int e = (int)(t >> 5);
    if (e >= E) return;
    int f = (int)(t & 31) << 2;
    int s = (int)src[e];
    int d = (int)dst[e];
    const float4 v = *reinterpret_cast<const float4*>(x + (unsigned)s * D_FEAT + f);
    float* a = agg + (unsigned)d * D_FEAT + f;
    atomicAdd(a + 0, v.x);
    atomicAdd(a + 1, v.y);
    atomicAdd(a + 2, v.z);
    atomicAdd(a + 3, v.w);
}

// ---------------------------------------------------------------- fused SAGE layer GEMM
// out[n, c] = act( (agg[n,:]/max(deg,1)) @ Wl[:,c] + bl[c] + x[n,:] @ Wr[:,c] )
// block = (32, NCOLS/16 tiles); one 16-row tile per block, one 16-col tile per wave.
template <int NCOLS, int RELU>
__global__ void sage_gemm(const float* __restrict__ agg, const float* __restrict__ deg,
                          const float* __restrict__ xin,
                          const float* __restrict__ Wl, const float* __restrict__ bl,
                          const float* __restrict__ Wr,
                          float* __restrict__ out, int n_nodes) {
    __shared__ float sA[16][132];   // agg tile, pre-scaled by 1/deg (stride 132 avoids 16-way bank clash)
    __shared__ float sX[16][132];   // root-feature tile
    __shared__ float sInv[16];

    const int m0 = blockIdx.x * 16;
    const int lane = threadIdx.x;                    // 0..31
    const int tid = threadIdx.y * 32 + lane;
    const int nthreads = blockDim.x * blockDim.y;

    if (tid < 16) {
        int row = m0 + tid;
        float dg = (row < n_nodes) ? deg[row] : 1.0f;
        sInv[tid] = 1.0f / fmaxf(dg, 1.0f);
    }
    __syncthreads();

    for (int i = tid; i < 16 * D_FEAT; i += nthreads) {
        int r = i >> 7;          // row within tile
        int cc = i & 127;        // feature
        int row = m0 + r;
        float av = 0.0f, xv = 0.0f;
        if (row < n_nodes) {
            unsigned off = (unsigned)row * D_FEAT + cc;   // 32-bit indexing
            av = agg[off] * sInv[r];
            xv = xin[off];
        }
        sA[r][cc] = av;
        sX[r][cc] = xv;
    }
    __syncthreads();

    // Per-lane fragment coordinates (ISA 7.12.2 layouts, wave32):
    //  A (16x4 f32): lane<16 -> M=lane, K={k,k+1}; lane>=16 -> M=lane-16, K={k+2,k+3}
    //  B (4x16 f32): mirror (lane%16 = N, lane/16 selects K pair)
    //  C (16x16 f32, 8 VGPRs): row = 8*(lane/16)+r, col = lane%16
    const int mr = lane & 15;
    const int koff = (lane >> 4) << 1;               // 0 or 2
    const int n0 = threadIdx.y * 16;
    const int col = n0 + mr;

    constexpr bool NEED_MASK = (NCOLS & 15) != 0;
    const float cmask = (!NEED_MASK || col < NCOLS) ? 1.0f : 0.0f;
    const int colc = (!NEED_MASK || col < NCOLS) ? col : (NCOLS - 1);

    const float* wl = Wl + (unsigned)koff * NCOLS + colc;   // lane-private weight base
    const float* wr = Wr + (unsigned)koff * NCOLS + colc;

    v8f c = {};

    #pragma unroll 8
    for (int k = 0; k < D_FEAT; k += 4) {
        v2f a, b;
        a[0] = sA[mr][k + koff];
        a[1] = sA[mr][k + koff + 1];
        b[0] = wl[(unsigned)(k)*NCOLS];
        b[1] = wl[(unsigned)(k + 1) * NCOLS];
        if (NEED_MASK) { b[0] *= cmask; b[1] *= cmask; }
        c = __builtin_amdgcn_wmma_f32_16x16x4_f32(false, a, false, b, (short)0, c, false, false);
    }
    #pragma unroll 8
    for (int k = 0; k < D_FEAT; k += 4) {
        v2f a, b;
        a[0] = sX[mr][k + koff];
        a[1] = sX[mr][k + koff + 1];
        b[0] = wr[(unsigned)(k)*NCOLS];
        b[1] = wr[(unsigned)(k + 1) * NCOLS];
        if (NEED_MASK) { b[0] *= cmask; b[1] *= cmask; }
        c = __builtin_amdgcn_wmma_f32_16x16x4_f32(false, a, false, b, (short)0, c, false, false);
    }

    float bias = bl[colc];
    if (NEED_MASK) bias *= cmask;
    const int rbase = m0 + ((lane >> 4) << 3);
    #pragma unroll
    for (int r = 0; r < 8; ++r) {
        int row = rbase + r;
        if (row < n_nodes && (!NEED_MASK || col < NCOLS)) {
            float v = c[r] + bias;
            if (RELU) v = fmaxf(v, 0.0f);
            out[(unsigned)row * NCOLS + col] = v;
        }
    }
}

// ---------------------------------------------------------------- L2-normalize + log_softmax
// one wave per node; lane covers classes {lane, lane+32}
__global__ void finalize_kernel(const float* __restrict__ h, float* __restrict__ out, int n) {
    int node = (int)(((size_t)blockIdx.x * blockDim.x + threadIdx.x) >> 5);
    int lane = threadIdx.x & 31;
    if (node >= n) return;

    const float* row = h + (unsigned)node * N_CLASSES;
    float v0 = (lane < N_CLASSES) ? row[lane] : 0.0f;
    float v1 = (lane + 32 < N_CLASSES) ? row[lane + 32] : 0.0f;

    float ss = v0 * v0 + v1 * v1;
    for (int o = 16; o; o >>= 1) ss += __shfl_xor(ss, o, 32);
    float inv = 1.0f / fmaxf(sqrtf(ss), 1e-12f);
    float u0 = v0 * inv, u1 = v1 * inv;

    float m0 = (lane < N_CLASSES) ? u0 : -INFINITY;
    float m1 = (lane + 32 < N_CLASSES) ? u1 : -INFINITY;
    float mx = fmaxf(m0, m1);
    for (int o = 16; o; o >>= 1) mx = fmaxf(mx, __shfl_xor(mx, o, 32));

    float e = ((lane < N_CLASSES) ? expf(u0 - mx) : 0.0f) +
              ((lane + 32 < N_CLASSES) ? expf(u1 - mx) : 0.0f);
    for (int o = 16; o; o >>= 1) e += __shfl_xor(e, o, 32);
    float lse = logf(e);

    if (lane < N_CLASSES) out[(unsigned)node * N_CLASSES + lane] = u0 - mx - lse;
    if (lane + 32 < N_CLASSES) out[(unsigned)node * N_CLASSES + lane + 32] = u1 - mx - lse;
}

// ---------------------------------------------------------------- launch
extern "C" void kernel_launch(void* const* d_in, const int* in_sizes, int n_in,
                              void* d_out, int out_size, void* d_ws, size_t ws_size,
                              hipStream_t stream) {
    const float* x = (const float*)d_in[0];
    const long long* ei = (const long long*)d_in[1];
    const long long* src = ei;                // edge_index[0]
    const long long* dst = ei + N_EDGES;      // edge_index[1]
    const float* Wl0 = (const float*)d_in[2];
    const float* bl0 = (const float*)d_in[3];
    const float* Wr0 = (const float*)d_in[4];
    const float* Wl1 = (const float*)d_in[5];
    const float* bl1 = (const float*)d_in[6];
    const float* Wr1 = (const float*)d_in[7];
    const float* Wl2 = (const float*)d_in[8];
    const float* bl2 = (const float*)d_in[9];
    const float* Wr2 = (const float*)d_in[10];
    float* out = (float*)d_out;

    // workspace layout (floats): agg[N*128] | deg[N] | hA[N*128] | hB[N*128] | h2[N*47]
    float* agg = (float*)d_ws;
    float* deg = agg + (size_t)N_NODES * D_FEAT;
    float* hA  = deg + N_NODES;
    float* hB  = hA + (size_t)N_NODES * D_FEAT;
    float* h2  = hB + (size_t)N_NODES * D_FEAT;

    const int rowTiles = (N_NODES + 15) / 16;                 // 3125
    const long long scatterThreads = (long long)N_EDGES * 32;
    const int scatterBlocks = (int)((scatterThreads + 255) / 256);

    // degree (layer-invariant) + layer 0 aggregate
    zero_f32<<<1024, 256, 0, stream>>>(deg, (size_t)N_NODES);
    zero_f32<<<2048, 256, 0, stream>>>(agg, (size_t)N_NODES * D_FEAT);
    deg_kernel<<<(N_EDGES + 255) / 256, 256, 0, stream>>>(dst, deg, N_EDGES);
    scatter_add<<<scatterBlocks, 256, 0, stream>>>(x, src, dst, agg, N_EDGES);
    sage_gemm<D_FEAT, 1><<<rowTiles, dim3(32, 8), 0, stream>>>(agg, deg, x, Wl0, bl0, Wr0, hA,
                                                               N_NODES);

    // layer 1
    zero_f32<<<2048, 256, 0, stream>>>(agg, (size_t)N_NODES * D_FEAT);
    scatter_add<<<scatterBlocks, 256, 0, stream>>>(hA, src, dst, agg, N_EDGES);
    sage_gemm<D_FEAT, 1><<<rowTiles, dim3(32, 8), 0, stream>>>(agg, deg, hA, Wl1, bl1, Wr1, hB,
                                                               N_NODES);

    // layer 2 (47 classes -> 3 col tiles, no relu)
    zero_f32<<<2048, 256, 0, stream>>>(agg, (size_t)N_NODES * D_FEAT);
    scatter_add<<<scatterBlocks, 256, 0, stream>>>(hB, src, dst, agg, N_EDGES);
    sage_gemm<N_CLASSES, 0><<<rowTiles, dim3(32, 3), 0, stream>>>(agg, deg, hB, Wl2, bl2, Wr2, h2,
                                                                  N_NODES);

    // normalize + log_softmax
    const long long finThreads = (long long)N_NODES * 32;
    finalize_kernel<<<(int)((finThreads + 255) / 256), 256, 0, stream>>>(h2, out, N_NODES);
}